// CPUNF4Embedding_78855599555225
// MI455X (gfx1250) — compile-verified
//
#include <hip/hip_runtime.h>
#include <stdint.h>

// NF4 embedding dequant-gather for MI455X (gfx1250).
// Per token: TDM-DMA one 8KB packed row into LDS, dequant via LDS 256-entry
// float2 LUT, write 16KB fp32 out with non-temporal 128-bit stores.

#define D_DIM  4096
#define HALF_D 2048   // int32 codes per row (one byte of payload each)

typedef __attribute__((ext_vector_type(4))) unsigned int u32x4;
typedef __attribute__((ext_vector_type(8))) int          i32x8;
typedef __attribute__((ext_vector_type(4))) int          i32x4;
typedef __attribute__((ext_vector_type(4))) float        f32x4;
typedef __attribute__((ext_vector_type(2))) float        f32x2;

__global__ __launch_bounds__(256) void nf4_embed_kernel(
    const int*   __restrict__ x,        // [ntok] row indices
    const int*   __restrict__ packed,   // [V, HALF_D] int32, value in [0,256)
    const float* __restrict__ nf4_lut,  // [16]
    const float* __restrict__ c,        // [1], out = lut[idx] / c
    float*       __restrict__ out,      // [ntok, D_DIM]
    int ntok)
{
    __shared__ __align__(16) int   row[HALF_D];   // 8 KB staging (TDM target)
    __shared__ f32x2               lut2[256];     // byte -> {hi,lo} dequant

    const int tok = blockIdx.x;
    if (tok >= ntok) return;
    const int t = threadIdx.x;

    // Wave-uniform row base address for this token's embedding row.
    const int rowidx = x[tok];
    const unsigned long long gaddr =
        (unsigned long long)(uintptr_t)packed +
        (unsigned long long)rowidx * (HALF_D * 4ull);

    // Tensor ops ignore EXEC: must skip via a *scalar* branch so only wave 0
    // issues the DMA.
    const unsigned wave =
        (unsigned)__builtin_amdgcn_readfirstlane((int)(threadIdx.x >> 5));
    if (wave == 0) {
        // ---- D# group 0 (128b): count=1, lds_addr, global_addr, type=2 ----
        u32x4 g0;
        g0[0] = 1u;                                        // count=1, user mode
        g0[1] = (unsigned)(uintptr_t)(&row[0]);            // LDS byte address
        g0[2] = (unsigned)(gaddr & 0xffffffffull);         // global_addr[31:0]
        g0[3] = (unsigned)((gaddr >> 32) & 0x01ffffffull)  // global_addr[56:32]
              | (2u << 30);                                // type = 2 ("image")
        // ---- D# group 1 (256b): dims/strides for a 2048x1 tile of 4B elems ----
        i32x8 g1;
        g1[0] = 0x00020000;  // data_size=2 (4-byte elements), no mask/pad/iter
        g1[1] = 0x08000000;  // tensor_dim0 = 2048 (low16 at bits 63:48)
        g1[2] = 0x00010000;  // tensor_dim0 hi16=0 | tensor_dim1 = 1
        g1[3] = 0x08000000;  // tensor_dim1 hi16=0 | tile_dim0 = 2048
        g1[4] = 0x00000001;  // tile_dim1 = 1, tile_dim2 = 0
        g1[5] = 0x00000800;  // tensor_dim0_stride = 2048 (low 32)
        g1[6] = 0x08000000;  // stride0 hi16=0 | tensor_dim1_stride lo16 = 2048
        g1[7] = 0;           // stride1 hi bits = 0
        i32x4 z4 = {0, 0, 0, 0};
#if defined(__clang_major__) && (__clang_major__ >= 23)
        i32x8 z8 = {0, 0, 0, 0, 0, 0, 0, 0};
        __builtin_amdgcn_tensor_load_to_lds(g0, g1, z4, z4, z8, 0);
#else
        __builtin_amdgcn_tensor_load_to_lds(g0, g1, z4, z4, 0);
#endif
    }

    // Build the dequant LUT while the DMA is in flight.
    // reference: out = nf4_lut[idx] / c  ->  prescale LUT by 1/c.
    {
        const float s  = 1.0f / c[0];
        const float hi = nf4_lut[(t >> 4) & 15];
        const float lo = nf4_lut[t & 15];
        f32x2 v; v.x = hi * s; v.y = lo * s;
        lut2[t] = v;
    }

    if (wave == 0) __builtin_amdgcn_s_wait_tensorcnt(0);
    __syncthreads();   // LUT stores + TDM-filled row visible to all waves

    // Each thread: codes [8t, 8t+8) -> floats [16t, 16t+16).
    const i32x4 a = ((const i32x4*)row)[2 * t + 0];
    const i32x4 b = ((const i32x4*)row)[2 * t + 1];

    f32x2 e;
    f32x4 o0, o1, o2, o3;
    e = lut2[a.x & 255]; o0.x = e.x; o0.y = e.y;
    e = lut2[a.y & 255]; o0.z = e.x; o0.w = e.y;
    e = lut2[a.z & 255]; o1.x = e.x; o1.y = e.y;
    e = lut2[a.w & 255]; o1.z = e.x; o1.w = e.y;
    e = lut2[b.x & 255]; o2.x = e.x; o2.y = e.y;
    e = lut2[b.y & 255]; o2.z = e.x; o2.w = e.y;
    e = lut2[b.z & 255]; o3.x = e.x; o3.y = e.y;
    e = lut2[b.w & 255]; o3.z = e.x; o3.w = e.y;

    // Output never re-read: non-temporal 128-bit stores keep L2 for the table.
    f32x4* dst = (f32x4*)(out + (size_t)tok * D_DIM + (size_t)t * 16);
    __builtin_nontemporal_store(o0, dst + 0);
    __builtin_nontemporal_store(o1, dst + 1);
    __builtin_nontemporal_store(o2, dst + 2);
    __builtin_nontemporal_store(o3, dst + 3);
}

extern "C" void kernel_launch(void* const* d_in, const int* in_sizes, int n_in,
                              void* d_out, int out_size, void* d_ws, size_t ws_size,
                              hipStream_t stream) {
    const int*   x       = (const int*)d_in[0];
    const int*   packed  = (const int*)d_in[1];
    const float* nf4_lut = (const float*)d_in[2];
    const float* c       = (const float*)d_in[3];
    float*       out     = (float*)d_out;

    const int ntok = in_sizes[0];  // B*S = 16384 tokens
    nf4_embed_kernel<<<ntok, 256, 0, stream>>>(x, packed, nf4_lut, c, out, ntok);
}